// RealAgnosticInteractionBlock_68418829025722
// MI455X (gfx1250) — compile-verified
//
#include <hip/hip_runtime.h>

// Problem constants (from reference)
#define NN      10000
#define EE      160000
#define CC      128
#define AA      10
#define SQRT3_INV 0.57735026918962576f

typedef float v8f __attribute__((ext_vector_type(8)));
typedef float v2f __attribute__((ext_vector_type(2)));

// ---------------------------------------------------------------------------
// WMMA helpers: D(16x16 f32) = A(16x4 f32) * B(4x16 f32) + C
//   A: lane L holds row (L&15), k = kk + 2*(L>>4) + {0,1}
//   B: lane L holds col (L&15), k = kk + 2*(L>>4) + {0,1}
//   C/D: reg r in lane L = D[r + 8*(L>>4)][L&15]
// ---------------------------------------------------------------------------
__device__ __forceinline__ v8f wmma4(v2f a, v2f b, v8f c) {
  return __builtin_amdgcn_wmma_f32_16x16x4_f32(
      false, a, false, b, (short)0, c, false, false);
}

__device__ __forceinline__ v2f ldsA(const float* buf, int ld, int kk, int lane) {
  int row = lane & 15, grp = lane >> 4;
  return *(const v2f*)(buf + row * ld + kk + 2 * grp);
}

// B fragment from fragment-major packed weights: one global_load_b64.
__device__ __forceinline__ v2f gBp(const float* __restrict__ P, int K4,
                                   int ct, int kk4, int lane) {
  return ((const v2f*)P)[(ct * K4 + kk4) * 32 + lane];
}

__device__ __forceinline__ void stC(float* buf, int ld, int coloff, v8f c, int lane) {
  int col = coloff + (lane & 15), grp = lane >> 4;
#pragma unroll
  for (int r = 0; r < 8; ++r) buf[(r + 8 * grp) * ld + col] = c[r];
}

__device__ __forceinline__ float silu(float x) { return x / (1.0f + __expf(-x)); }
__device__ __forceinline__ float sigmoidf(float x) { return 1.0f / (1.0f + __expf(-x)); }

// ---------------------------------------------------------------------------
// Software-pipelined GEMM core: K4tot (even) k-steps, NCT output col tiles.
// Double-buffered A + B banks so next-stage loads overlap current WMMAs.
// ---------------------------------------------------------------------------
template<int NCT, typename AL>
__device__ __forceinline__ void gemm_pipe(int K4tot, AL loadA,
    const float* __restrict__ P, int K4ld, int ct0, v8f* acc, int lane) {
  v2f a0, a1, b0[NCT], b1[NCT];
  a0 = loadA(0);
#pragma unroll
  for (int ct = 0; ct < NCT; ++ct) b0[ct] = gBp(P, K4ld, ct0 + ct, 0, lane);
  for (int k = 0; k + 2 < K4tot; k += 2) {
    a1 = loadA(k + 1);
#pragma unroll
    for (int ct = 0; ct < NCT; ++ct) b1[ct] = gBp(P, K4ld, ct0 + ct, k + 1, lane);
#pragma unroll
    for (int ct = 0; ct < NCT; ++ct) acc[ct] = wmma4(a0, b0[ct], acc[ct]);
    a0 = loadA(k + 2);
#pragma unroll
    for (int ct = 0; ct < NCT; ++ct) b0[ct] = gBp(P, K4ld, ct0 + ct, k + 2, lane);
#pragma unroll
    for (int ct = 0; ct < NCT; ++ct) acc[ct] = wmma4(a1, b1[ct], acc[ct]);
  }
  a1 = loadA(K4tot - 1);
#pragma unroll
  for (int ct = 0; ct < NCT; ++ct) b1[ct] = gBp(P, K4ld, ct0 + ct, K4tot - 1, lane);
#pragma unroll
  for (int ct = 0; ct < NCT; ++ct) acc[ct] = wmma4(a0, b0[ct], acc[ct]);
#pragma unroll
  for (int ct = 0; ct < NCT; ++ct) acc[ct] = wmma4(a1, b1[ct], acc[ct]);
}

// ---------------------------------------------------------------------------
__global__ void zero_kernel(float* __restrict__ p, int n) {
  int i = blockIdx.x * blockDim.x + threadIdx.x;
  if (i < n) p[i] = 0.0f;
}

// ---------------------------------------------------------------------------
// Weight repack into fragment-major order (B fragment = one b64 load).
// ---------------------------------------------------------------------------
struct RepackDesc { const float* W; float* out; int K4; int NT; int rs; };
struct RepackArgs { RepackDesc d[12]; };

__global__ void repack_all(RepackArgs args) {
  RepackDesc dd = args.d[blockIdx.y];
  int i = blockIdx.x * blockDim.x + threadIdx.x;
  if (i >= dd.K4 * dd.NT * 32) return;
  int lane = i & 31;
  int fi = i >> 5;                 // ct*K4 + kk4
  int ct = fi / dd.K4, kk4 = fi - ct * dd.K4;
  int row0 = kk4 * 4 + 2 * (lane >> 4);
  int col = ct * 16 + (lane & 15);
  dd.out[i * 2 + 0] = dd.W[row0 * dd.rs + col];
  dd.out[i * 2 + 1] = dd.W[(row0 + 1) * dd.rs + col];
}

// ---------------------------------------------------------------------------
// Kernel 1: node up-projection. One wave (block=32), grid = (625, 4).
// ---------------------------------------------------------------------------
__global__ void up_kernel(const float* __restrict__ node_feats,
                          const float* __restrict__ P_up0,
                          const float* __restrict__ P_up1,
                          float* __restrict__ s_up,
                          float* __restrict__ v_up) {
  int lane = threadIdx.x & 31;
  int nt = blockIdx.x, mat = blockIdx.y;   // uniform
  int row = lane & 15, grp = lane >> 4;
  int n = nt * 16 + row;
  const float* P = (mat == 0) ? P_up0 : P_up1;

  v8f acc8[8] = {};
  if (mat == 0) {
    const float* base = node_feats + n * 512 + 2 * grp;
    gemm_pipe<8>(32, [&](int k) { return *(const v2f*)(base + k * 4); },
                 P, 32, 0, acc8, lane);
  } else {
    const float* base = node_feats + n * 512 + 128 + (mat - 1) + 2 * grp * 3;
    gemm_pipe<8>(32, [&](int k) {
                   v2f a; a.x = base[k * 12]; a.y = base[k * 12 + 3]; return a;
                 }, P, 32, 0, acc8, lane);
  }
  const float f = 0.08838834764831845f;  // 1/sqrt(128)
  for (int ct = 0; ct < 8; ++ct) {
#pragma unroll
    for (int r = 0; r < 8; ++r) {
      int nn = nt * 16 + r + 8 * grp;
      int c = ct * 16 + (lane & 15);
      float v = acc8[ct][r] * f;
      if (mat == 0) s_up[nn * 128 + c] = v;
      else          v_up[(nn * 128 + c) * 3 + (mat - 1)] = v;
    }
  }
}

// ---------------------------------------------------------------------------
// mji regions (never straddle 16-wide feature tiles):
//   REG 0: feat in [0,128)    m1 = tpw0*ss*y0
//   REG 1: feat in [128,256)  m4 = tpw3*(vv.y1)/sqrt3
//   REG 2: feat in [256,640)  m2 = tpw1*ss*y1[m]     (q=feat-256, u=q/3, m=q%3)
//   REG 3: feat in [640,1024) m3 = tpw2*vv[m]*y0     (q=feat-640, c=q/3, m=q%3)
// Pass-1 tile fill is split: prefetch_tile issues the global gathers into
// registers; combine_store folds in LDS operands and writes the mbuf tile.
// Fill mapping: value t of a lane -> el = 2t+grp, j = lane&15.
// ---------------------------------------------------------------------------
template<int REG>
__device__ __forceinline__ void prefetch_tile(int ft, float (*g)[3],
    const int* msrc, const float* __restrict__ s_up,
    const float* __restrict__ v_up, int lane) {
  int j = lane & 15;
#pragma unroll
  for (int t = 0; t < 8; ++t) {
    int feat = ft * 16 + j;
    int src = msrc[t];
    if (REG == 0) {
      g[t][0] = s_up[src * 128 + feat];
    } else if (REG == 1) {
      int c = feat - 128;
      const float* vv = v_up + (src * 128 + c) * 3;
      g[t][0] = vv[0]; g[t][1] = vv[1]; g[t][2] = vv[2];
    } else if (REG == 2) {
      int q = feat - 256; int u = q / 3;
      g[t][0] = s_up[src * 128 + u];
    } else {
      int q = feat - 640; int c = q / 3, m = q - 3 * c;
      g[t][0] = v_up[(src * 128 + c) * 3 + m];
    }
  }
}

template<int REG>
__device__ __forceinline__ void combine_store_tile(int ft, const float (*g)[3],
    float* mb, const float* tpw, const float* eatt, int lane) {
  int j = lane & 15, grp = lane >> 4;
#pragma unroll
  for (int t = 0; t < 8; ++t) {
    int el = 2 * t + grp;
    int feat = ft * 16 + j;
    const float* ea = eatt + el * 4;
    const float* tp = tpw + el * 512;
    float val;
    if (REG == 0) {
      val = tp[feat] * g[t][0] * ea[0];
    } else if (REG == 1) {
      int c = feat - 128;
      float d = g[t][0] * ea[1] + g[t][1] * ea[2] + g[t][2] * ea[3];
      val = tp[384 + c] * d * SQRT3_INV;
    } else if (REG == 2) {
      int q = feat - 256; int u = q / 3, m = q - 3 * u;
      val = tp[128 + u] * g[t][0] * ea[1 + m];
    } else {
      int q = feat - 640; int c = q / 3;
      val = tp[256 + c] * g[t][0] * ea[0];
    }
    mb[el * 16 + j] = val;
  }
}

template<int REG>
__device__ __forceinline__ void pass1_region(int ft0, int ft1,
    float* mbA, float* mbB, v8f& g1a, v8f& g1b, const float* tpw,
    const int* msrc, const float* eatt,
    const float* __restrict__ s_up, const float* __restrict__ v_up,
    const float* __restrict__ P_se1, int lane) {
  float g[8][3];
  prefetch_tile<REG>(ft0, g, msrc, s_up, v_up, lane);
  combine_store_tile<REG>(ft0, g, mbA, tpw, eatt, lane);
  for (int ft = ft0; ft < ft1; ++ft) {
    float* cur = ((ft - ft0) & 1) ? mbB : mbA;
    float* nxt = ((ft - ft0) & 1) ? mbA : mbB;
    if (ft + 1 < ft1) prefetch_tile<REG>(ft + 1, g, msrc, s_up, v_up, lane);
    // batched B loads, then GEMM of current tile (overlaps in-flight gathers)
    v2f b[8];
#pragma unroll
    for (int kk4 = 0; kk4 < 4; ++kk4) {
      b[kk4 * 2]     = gBp(P_se1, 256, 0, ft * 4 + kk4, lane);
      b[kk4 * 2 + 1] = gBp(P_se1, 256, 1, ft * 4 + kk4, lane);
    }
#pragma unroll
    for (int kk4 = 0; kk4 < 4; ++kk4) {
      v2f a = ldsA(cur, 16, kk4 * 4, lane);
      g1a = wmma4(a, b[kk4 * 2], g1a);
      g1b = wmma4(a, b[kk4 * 2 + 1], g1b);
    }
    if (ft + 1 < ft1) combine_store_tile<REG>(ft + 1, g, nxt, tpw, eatt, lane);
  }
}

// ---------------------------------------------------------------------------
// Pass 2: gate + apply + scatter. ga[8] (A of gate GEMM) is loop-invariant.
// B banks double-buffered (region lengths are even). Planar accumulator:
// Acc[plane][NN][256], plane 0 = scalar, 1..3 = vector components.
// ---------------------------------------------------------------------------
template<int REG>
__device__ __forceinline__ void apply_tile(int ft, v8f g2,
    const float* tpw, const int* esrc, const int* edst, const float* eatt,
    const float* __restrict__ s_up, const float* __restrict__ v_up,
    float* __restrict__ Acc, int lane) {
  int grp = lane >> 4, j = lane & 15;
#pragma unroll
  for (int r = 0; r < 8; ++r) {
    int el = r + 8 * grp;
    int feat = ft * 16 + j;
    float gate = sigmoidf(g2[r]);
    int src = esrc[el], dst = edst[el];
    const float* ea = eatt + el * 4;
    const float* tp = tpw + el * 512;
    if (REG == 0) {
      float val = tp[feat] * s_up[src * 128 + feat] * ea[0] * gate;
      atomicAdd(&Acc[dst * 256 + feat], val);
    } else if (REG == 1) {
      int c = feat - 128;
      const float* vv = v_up + (src * 128 + c) * 3;
      float d = vv[0] * ea[1] + vv[1] * ea[2] + vv[2] * ea[3];
      atomicAdd(&Acc[dst * 256 + feat], tp[384 + c] * d * SQRT3_INV * gate);
    } else if (REG == 2) {
      int q = feat - 256; int u = q / 3, m = q - 3 * u;
      float val = tp[128 + u] * s_up[src * 128 + u] * ea[1 + m] * gate;
      atomicAdd(&Acc[((1 + m) * NN + dst) * 256 + u], val);
    } else {
      int q = feat - 640; int c = q / 3, m = q - 3 * c;
      float val = tp[256 + c] * v_up[(src * 128 + c) * 3 + m] * ea[0] * gate;
      atomicAdd(&Acc[((1 + m) * NN + dst) * 256 + 128 + c], val);
    }
  }
}

template<int REG>
__device__ __forceinline__ void pass2_region(int ft0, int ft1, const v2f* ga,
    const float* tpw, const int* esrc, const int* edst, const float* eatt,
    const float* __restrict__ s_up, const float* __restrict__ v_up,
    const float* __restrict__ P_se2, float* __restrict__ Acc, int lane) {
  v2f b0[8], b1[8];
#pragma unroll
  for (int kk4 = 0; kk4 < 8; ++kk4) b0[kk4] = gBp(P_se2, 8, ft0, kk4, lane);
  for (int ft = ft0; ft < ft1; ft += 2) {
#pragma unroll
    for (int kk4 = 0; kk4 < 8; ++kk4) b1[kk4] = gBp(P_se2, 8, ft + 1, kk4, lane);
    {
      v8f g2 = {};
#pragma unroll
      for (int kk4 = 0; kk4 < 8; ++kk4) g2 = wmma4(ga[kk4], b0[kk4], g2);
      apply_tile<REG>(ft, g2, tpw, esrc, edst, eatt, s_up, v_up, Acc, lane);
    }
    if (ft + 2 < ft1) {
#pragma unroll
      for (int kk4 = 0; kk4 < 8; ++kk4) b0[kk4] = gBp(P_se2, 8, ft + 2, kk4, lane);
    }
    {
      v8f g2 = {};
#pragma unroll
      for (int kk4 = 0; kk4 < 8; ++kk4) g2 = wmma4(ga[kk4], b1[kk4], g2);
      apply_tile<REG>(ft + 1, g2, tpw, esrc, edst, eatt, s_up, v_up, Acc, lane);
    }
  }
}

// ---------------------------------------------------------------------------
// Kernel 2: fused edge pipeline. One wave = 16 edges, block = 64 (2 waves).
// Per-wave LDS (floats): tpw 8192 | hA 1024 | hB 1024 | mbA 256 | mbB 256 |
//                        gbuf 512 | eatt 64 | esrc 16 | edst 16 = 11360
// ---------------------------------------------------------------------------
__global__ void edge_kernel(const float* __restrict__ edge_attrs,
                            const float* __restrict__ edge_feats,
                            const int* __restrict__ edge_index,
                            const float* __restrict__ P_r1,
                            const float* __restrict__ P_r2,
                            const float* __restrict__ P_r3,
                            const float* __restrict__ P_r4,
                            const float* __restrict__ P_se1,
                            const float* __restrict__ P_se2,
                            const float* __restrict__ s_up,
                            const float* __restrict__ v_up,
                            float* __restrict__ Acc) {
  extern __shared__ float smem[];
  int lane = threadIdx.x & 31, wave = threadIdx.x >> 5;
  float* ws   = smem + wave * 11360;
  float* tpw  = ws;              // 8192
  float* hA   = ws + 8192;       // 1024
  float* hB   = ws + 9216;       // 1024
  float* mbA  = ws + 10240;      // 256
  float* mbB  = ws + 10496;      // 256
  float* gbuf = ws + 10752;      // 512
  float* eatt = ws + 11264;      // 64
  int*   esrc = (int*)(ws + 11328);  // 16
  int*   edst = (int*)(ws + 11344);  // 16

  int e0 = (blockIdx.x * 2 + wave) * 16;
  int row = lane & 15, grp = lane >> 4;

  if (lane < 16) {
    esrc[lane] = edge_index[e0 + lane];
    edst[lane] = edge_index[EE + e0 + lane];
  }
#pragma unroll
  for (int t = 0; t < 2; ++t) eatt[t * 32 + lane] = edge_attrs[e0 * 4 + t * 32 + lane];

  const float F1 = 0.35355339059327373f;  // 1/sqrt(8)
  const float F64 = 0.125f;               // 1/sqrt(64)

  // --- layer 1: 16x8 @ 8x64 -> hA ------------------------------------------
  {
    const float* efp = edge_feats + (e0 + row) * 8 + 2 * grp;
    v8f acc4[4] = {};
    gemm_pipe<4>(2, [&](int k) { return *(const v2f*)(efp + k * 4); },
                 P_r1, 2, 0, acc4, lane);
    for (int ct = 0; ct < 4; ++ct) {
      v8f o;
#pragma unroll
      for (int r = 0; r < 8; ++r) o[r] = silu(acc4[ct][r] * F1);
      stC(hA, 64, ct * 16, o, lane);
    }
  }
  // --- layer 2: hA @ Wr2 -> hB ---------------------------------------------
  {
    const float* hp = hA + row * 64 + 2 * grp;
    v8f acc4[4] = {};
    gemm_pipe<4>(16, [&](int k) { return *(const v2f*)(hp + k * 4); },
                 P_r2, 16, 0, acc4, lane);
    for (int ct = 0; ct < 4; ++ct) {
      v8f o;
#pragma unroll
      for (int r = 0; r < 8; ++r) o[r] = silu(acc4[ct][r] * F64);
      stC(hB, 64, ct * 16, o, lane);
    }
  }
  // --- layer 3: hB @ Wr3 -> hA ---------------------------------------------
  {
    const float* hp = hB + row * 64 + 2 * grp;
    v8f acc4[4] = {};
    gemm_pipe<4>(16, [&](int k) { return *(const v2f*)(hp + k * 4); },
                 P_r3, 16, 0, acc4, lane);
    for (int ct = 0; ct < 4; ++ct) {
      v8f o;
#pragma unroll
      for (int r = 0; r < 8; ++r) o[r] = silu(acc4[ct][r] * F64);
      stC(hA, 64, ct * 16, o, lane);
    }
  }
  // --- layer 4: hA @ Wr4 -> tpw (16x512); 4 groups of 8 col tiles ----------
  {
    const float* hp = hA + row * 64 + 2 * grp;
    for (int gct = 0; gct < 4; ++gct) {
      v8f acc8[8] = {};
      gemm_pipe<8>(16, [&](int k) { return *(const v2f*)(hp + k * 4); },
                   P_r4, 16, gct * 8, acc8, lane);
      for (int ct = 0; ct < 8; ++ct) {
        v8f o;
#pragma unroll
        for (int r = 0; r < 8; ++r) o[r] = acc8[ct][r] * F64;
        stC(tpw, 512, (gct * 8 + ct) * 16, o, lane);
      }
    }
  }

  // register-cached source indices for pass-1 fill mapping (el = 2t+grp)
  int msrc[8];
#pragma unroll
  for (int t = 0; t < 8; ++t) msrc[t] = esrc[2 * t + grp];

  // --- pass 1: g1 = mji @ W_se1 (16x32) ------------------------------------
  v8f g1a = {}, g1b = {};
  pass1_region<0>(0, 8,   mbA, mbB, g1a, g1b, tpw, msrc, eatt, s_up, v_up, P_se1, lane);
  pass1_region<1>(8, 16,  mbA, mbB, g1a, g1b, tpw, msrc, eatt, s_up, v_up, P_se1, lane);
  pass1_region<2>(16, 40, mbA, mbB, g1a, g1b, tpw, msrc, eatt, s_up, v_up, P_se1, lane);
  pass1_region<3>(40, 64, mbA, mbB, g1a, g1b, tpw, msrc, eatt, s_up, v_up, P_se1, lane);

  // silu(g1) -> gbuf (LDS transpose) -> 8 A fragments in registers
  v8f sa, sb;
#pragma unroll
  for (int r = 0; r < 8; ++r) { sa[r] = silu(g1a[r]); sb[r] = silu(g1b[r]); }
  stC(gbuf, 32, 0, sa, lane);
  stC(gbuf, 32, 16, sb, lane);
  v2f ga[8];
#pragma unroll
  for (int kk4 = 0; kk4 < 8; ++kk4) ga[kk4] = ldsA(gbuf, 32, kk4 * 4, lane);

  // --- pass 2: gate + apply + scatter --------------------------------------
  pass2_region<0>(0, 8,   ga, tpw, esrc, edst, eatt, s_up, v_up, P_se2, Acc, lane);
  pass2_region<1>(8, 16,  ga, tpw, esrc, edst, eatt, s_up, v_up, P_se2, Acc, lane);
  pass2_region<2>(16, 40, ga, tpw, esrc, edst, eatt, s_up, v_up, P_se2, Acc, lane);
  pass2_region<3>(40, 64, ga, tpw, esrc, edst, eatt, s_up, v_up, P_se2, Acc, lane);
}

// ---------------------------------------------------------------------------
// Kernel 3: node output. One wave (block=32), grid = (625, 4).
// ---------------------------------------------------------------------------
__global__ void node_out_kernel(const float* __restrict__ node_attrs,
                                const float* __restrict__ Acc,
                                const float* __restrict__ P_lin0,
                                const float* __restrict__ P_lin1,
                                const float* __restrict__ P_sk0,
                                const float* __restrict__ P_sk1,
                                float* __restrict__ out) {
  extern __shared__ float smem[];
  float* tbuf = smem;                  // 16*128
  int lane = threadIdx.x & 31;
  int nt = blockIdx.x, mat = blockIdx.y;   // uniform
  int row = lane & 15, grp = lane >> 4;
  const float* plane = Acc + (size_t)mat * NN * 256;
  const float* Plin = (mat == 0) ? P_lin0 : P_lin1;
  const float* Psk  = (mat == 0) ? P_sk0  : P_sk1;

  const float flin = (1.0f / 16.0f) * (1.0f / 16.0f); // 1/sqrt(256) / AVG_NEIGH
  const float fsk  = 0.02795084971874737f;            // 1/sqrt(1280)

  int n = nt * 16 + row;
  // --- T = plane @ Wlin * flin -> tbuf (16x128) ----------------------------
  {
    const float* base = plane + n * 256 + 2 * grp;
    v8f acc8[8] = {};
    gemm_pipe<8>(64, [&](int k) { return *(const v2f*)(base + k * 4); },
                 Plin, 64, 0, acc8, lane);
    for (int ct = 0; ct < 8; ++ct) {
      v8f o;
#pragma unroll
      for (int r = 0; r < 8; ++r) o[r] = acc8[ct][r] * flin;
      stC(tbuf, 128, ct * 16, o, lane);
    }
  }

  float attr[AA];
#pragma unroll
  for (int w = 0; w < AA; ++w) attr[w] = node_attrs[(nt * 16 + row) * AA + w];

  // --- Out = sum_w diag(attr_w) T @ Wsk[:,w,:] -----------------------------
  {
    const float* tb = tbuf + row * 128 + 2 * grp;
    v8f acc8[8] = {};
    for (int w = 0; w < AA; ++w) {
      float aw = attr[w];
      gemm_pipe<8>(32, [&](int k) {
                     v2f a = *(const v2f*)(tb + k * 4);
                     a.x *= aw; a.y *= aw; return a;
                   }, Psk, 32, w * 8, acc8, lane);
    }
    for (int ct = 0; ct < 8; ++ct) {
#pragma unroll
      for (int r = 0; r < 8; ++r) {
        int nn = nt * 16 + r + 8 * grp;
        int v = ct * 16 + (lane & 15);
        out[(nn * 128 + v) * 4 + mat] = acc8[ct][r] * fsk;
      }
    }
  }
}

// ---------------------------------------------------------------------------
extern "C" void kernel_launch(void* const* d_in, const int* in_sizes, int n_in,
                              void* d_out, int out_size, void* d_ws, size_t ws_size,
                              hipStream_t stream) {
  const float* node_attrs = (const float*)d_in[0];
  const float* node_feats = (const float*)d_in[1];
  const float* edge_attrs = (const float*)d_in[2];
  const float* edge_feats = (const float*)d_in[3];
  const int*   edge_index = (const int*)d_in[4];
  const float* W_up0 = (const float*)d_in[5];
  const float* W_up1 = (const float*)d_in[6];
  const float* Wr1   = (const float*)d_in[7];
  const float* Wr2   = (const float*)d_in[8];
  const float* Wr3   = (const float*)d_in[9];
  const float* Wr4   = (const float*)d_in[10];
  const float* Wse1  = (const float*)d_in[11];
  const float* Wse2  = (const float*)d_in[12];
  const float* W_lin0 = (const float*)d_in[13];
  const float* W_lin1 = (const float*)d_in[14];
  const float* W_sk0  = (const float*)d_in[15];
  const float* W_sk1  = (const float*)d_in[16];
  (void)in_sizes; (void)n_in; (void)out_size; (void)ws_size;

  float* ws   = (float*)d_ws;
  float* s_up = ws;                       // N*128
  float* v_up = s_up + NN * 128;          // N*128*3 (interleaved xyz)
  float* Acc  = v_up + NN * 128 * 3;      // 4 planes of N*256
  float* pw   = Acc + NN * 1024;          // packed weights
  float* P_up0 = pw;            pw += 16384;
  float* P_up1 = pw;            pw += 16384;
  float* P_r1  = pw;            pw += 512;
  float* P_r2  = pw;            pw += 4096;
  float* P_r3  = pw;            pw += 4096;
  float* P_r4  = pw;            pw += 32768;
  float* P_se1 = pw;            pw += 32768;
  float* P_se2 = pw;            pw += 32768;
  float* P_lin0 = pw;           pw += 32768;
  float* P_lin1 = pw;           pw += 32768;
  float* P_sk0 = pw;            pw += 163840;
  float* P_sk1 = pw;            pw += 163840;

  int nzero = NN * 1024;
  zero_kernel<<<(nzero + 255) / 256, 256, 0, stream>>>(Acc, nzero);

  RepackArgs ra;
  ra.d[0]  = {W_up0,  P_up0,  32,  8,  128};
  ra.d[1]  = {W_up1,  P_up1,  32,  8,  128};
  ra.d[2]  = {Wr1,    P_r1,    2,  4,   64};
  ra.d[3]  = {Wr2,    P_r2,   16,  4,   64};
  ra.d[4]  = {Wr3,    P_r3,   16,  4,   64};
  ra.d[5]  = {Wr4,    P_r4,   16, 32,  512};
  ra.d[6]  = {Wse1,   P_se1, 256,  2,   32};
  ra.d[7]  = {Wse2,   P_se2,   8, 64, 1024};
  ra.d[8]  = {W_lin0, P_lin0, 64,  8,  128};
  ra.d[9]  = {W_lin1, P_lin1, 64,  8,  128};
  ra.d[10] = {W_sk0,  P_sk0,  32, 80, 1280};
  ra.d[11] = {W_sk1,  P_sk1,  32, 80, 1280};
  repack_all<<<dim3(320, 12), 256, 0, stream>>>(ra);

  up_kernel<<<dim3(625, 4), 32, 0, stream>>>(node_feats, P_up0, P_up1, s_up, v_up);

  edge_kernel<<<EE / 32, 64, 2 * 11360 * sizeof(float), stream>>>(
      edge_attrs, edge_feats, edge_index, P_r1, P_r2, P_r3, P_r4, P_se1, P_se2,
      s_up, v_up, Acc);

  node_out_kernel<<<dim3(625, 4), 32, 16 * 128 * sizeof(float), stream>>>(
      node_attrs, Acc, P_lin0, P_lin1, P_sk0, P_sk1, (float*)d_out);
}